// Net_31576599560690
// MI455X (gfx1250) — compile-verified
//
#include <hip/hip_runtime.h>

#define N_NODES  100000
#define N_EDGES  1600000
#define DIM      64
#define N_GRAPHS 128
#define N_LAYERS 4
#define BN_EPS   1e-5f

typedef __attribute__((ext_vector_type(2))) float v2f;
typedef __attribute__((ext_vector_type(8))) float v8f;

// ---------------------------------------------------------------------------
// agg = (1 + eps[l]) * h          (eps read from device memory: capture-safe)
// ---------------------------------------------------------------------------
__global__ void k_scale(const float* __restrict__ h, float* __restrict__ agg,
                        const float* __restrict__ eps, int l, int n) {
    int i = blockIdx.x * blockDim.x + threadIdx.x;
    if (i < n) agg[i] = (1.0f + eps[l]) * h[i];
}

// ---------------------------------------------------------------------------
// GIN sum aggregation: agg[dst,:] += h[src,:].  4 threads per edge, each
// thread moves 16 floats (4x float4 gather, 16 global_atomic_add_f32).
// ---------------------------------------------------------------------------
__global__ void k_scatter(const float* __restrict__ h, const int* __restrict__ src,
                          const int* __restrict__ dst, float* __restrict__ agg) {
    long long gid = (long long)blockIdx.x * blockDim.x + threadIdx.x;
    if (gid >= (long long)N_EDGES * 4) return;
    int e = (int)(gid >> 2);
    int q = ((int)gid & 3) * 16;
    int s = src[e], d = dst[e];
    const float* hp = h + (long long)s * DIM + q;
    float* p = agg + (long long)d * DIM + q;
#pragma unroll
    for (int j = 0; j < 4; ++j) {
        float4 v = *(const float4*)(hp + 4 * j);
        atomicAdd(p + 4 * j + 0, v.x);
        atomicAdd(p + 4 * j + 1, v.y);
        atomicAdd(p + 4 * j + 2, v.z);
        atomicAdd(p + 4 * j + 3, v.w);
    }
}

// ---------------------------------------------------------------------------
// Y[N,64] = f(A)[N,64] @ W[64,64] + bias, fp32 WMMA (V_WMMA_F32_16X16X4_F32).
// One wave computes a full 16x64 output strip (4 accumulators): each A
// fragment is loaded once and reused across 4 column-tile WMMAs.
// APPLY_BN is a template parameter so the hot loop has no per-iteration
// uniform branch; BN path: a = relu(a*scale[k]+shift[k]) fused into A load.
// Fragment layouts per CDNA5 ISA 7.12.2:
//   A 16x4 : lane L holds row m=L&15, K = (L>>4)*2 + {0,1}
//   B 4x16 : lane L holds col n=L&15, K = (L>>4)*2 + {0,1}
//   C 16x16: VGPR r holds row r + 8*(L>>4), col n=L&15
// ---------------------------------------------------------------------------
template <int APPLY_BN>
__global__ void k_gemm64(const float* __restrict__ A, const float* __restrict__ W,
                         const float* __restrict__ bias, const float* __restrict__ ss,
                         float* __restrict__ Y) {
    int wave  = threadIdx.x >> 5;
    int lane  = threadIdx.x & 31;
    int strip = blockIdx.x * 8 + wave;          // 6250 strips of 16 rows
    if (strip >= N_NODES / 16) return;          // wave-uniform: EXEC stays all-1
    int row0 = strip * 16;
    int m    = lane & 15;
    int sel  = lane >> 4;                       // 0 or 1

    v8f acc[4] = {};
    const float* arow = A + (long long)(row0 + m) * DIM + sel * 2;
#pragma unroll
    for (int k0 = 0; k0 < DIM; k0 += 4) {
        int kb = k0 + sel * 2;
        v2f a;
        a.x = arow[k0];
        a.y = arow[k0 + 1];
        if (APPLY_BN) {                         // fused BN1 + ReLU on the input
            float y0 = a.x * ss[kb]     + ss[64 + kb];
            float y1 = a.y * ss[kb + 1] + ss[64 + kb + 1];
            a.x = y0 > 0.0f ? y0 : 0.0f;
            a.y = y1 > 0.0f ? y1 : 0.0f;
        }
        const float* w0 = W + (kb + 0) * DIM + m;
        const float* w1 = W + (kb + 1) * DIM + m;
#pragma unroll
        for (int t = 0; t < 4; ++t) {
            v2f b;
            b.x = w0[16 * t];
            b.y = w1[16 * t];
            acc[t] = __builtin_amdgcn_wmma_f32_16x16x4_f32(
                /*neg_a=*/false, a, /*neg_b=*/false, b,
                /*c_mod=*/(short)0, acc[t], /*reuse_a=*/false, /*reuse_b=*/false);
        }
    }
#pragma unroll
    for (int t = 0; t < 4; ++t) {
        int col  = 16 * t + m;
        float bv = bias[col];
#pragma unroll
        for (int r = 0; r < 8; ++r)
            Y[(long long)(row0 + r + 8 * sel) * DIM + col] = acc[t][r] + bv;
    }
}

// ---------------------------------------------------------------------------
// BatchNorm (training mode) column statistics over N rows, two-pass.
// stats[0:64]=sum, stats[64:128]=sumsq, stats[128:192]=scale, stats[192:256]=shift
// ---------------------------------------------------------------------------
__global__ void k_zero_stats(float* __restrict__ stats) {
    stats[threadIdx.x] = 0.0f;                  // 128 threads
}

__global__ void k_colstats(const float* __restrict__ X, float* __restrict__ stats) {
    __shared__ float ssum[256], ssq[256];
    int d    = threadIdx.x & 63;
    int rgrp = threadIdx.x >> 6;                // 4 row-groups per block
    float s = 0.0f, q = 0.0f;
    for (int row = blockIdx.x * 4 + rgrp; row < N_NODES; row += gridDim.x * 4) {
        float v = X[(long long)row * DIM + d];
        s += v; q += v * v;
    }
    ssum[threadIdx.x] = s; ssq[threadIdx.x] = q;
    __syncthreads();
    if (threadIdx.x < 64) {
        s = ssum[d] + ssum[d + 64] + ssum[d + 128] + ssum[d + 192];
        q = ssq[d]  + ssq[d + 64]  + ssq[d + 128]  + ssq[d + 192];
        atomicAdd(&stats[d], s);
        atomicAdd(&stats[64 + d], q);
    }
}

__global__ void k_bn_finalize(const float* __restrict__ stats,
                              const float* __restrict__ g, const float* __restrict__ b,
                              float* __restrict__ ss) {
    int d = threadIdx.x;                        // 64 threads
    float mean = stats[d] * (1.0f / (float)N_NODES);
    float var  = stats[64 + d] * (1.0f / (float)N_NODES) - mean * mean;
    float rstd = rsqrtf(var + BN_EPS);
    float sc   = g[d] * rstd;
    ss[d]      = sc;                            // scale  (stats+128)
    ss[64 + d] = b[d] - mean * sc;              // shift  (stats+192)
}

__global__ void k_bn_relu(float* __restrict__ X, const float* __restrict__ ss, int n) {
    int i = blockIdx.x * blockDim.x + threadIdx.x;
    if (i >= n) return;
    int d = i & 63;
    float v = X[i] * ss[d] + ss[64 + d];
    X[i] = v > 0.0f ? v : 0.0f;
}

// ---------------------------------------------------------------------------
// Fused pooling + 64x1 predictor: out[graph_ids[i]] += dot(h[i,:], pw)
// ---------------------------------------------------------------------------
__global__ void k_pool(const float* __restrict__ h, const int* __restrict__ gids,
                       const float* __restrict__ pw, float* __restrict__ out) {
    int i = blockIdx.x * blockDim.x + threadIdx.x;
    if (i >= N_NODES) return;
    const float* row = h + (long long)i * DIM;
    float acc = 0.0f;
#pragma unroll
    for (int d = 0; d < DIM; ++d) acc += row[d] * pw[d];
    atomicAdd(&out[gids[i]], acc);
}

__global__ void k_init_out(const float* __restrict__ pb, float* __restrict__ out) {
    float s = 0.0f;
#pragma unroll
    for (int l = 0; l <= N_LAYERS; ++l) s += pb[l];
    out[threadIdx.x] = s;                       // 128 threads
}

// ---------------------------------------------------------------------------
extern "C" void kernel_launch(void* const* d_in, const int* in_sizes, int n_in,
                              void* d_out, int out_size, void* d_ws, size_t ws_size,
                              hipStream_t stream) {
    const float* feat   = (const float*)d_in[0];
    const int*   esrc   = (const int*)  d_in[1];
    const int*   edst   = (const int*)  d_in[2];
    const int*   gids   = (const int*)  d_in[3];
    const float* eps    = (const float*)d_in[4];
    const float* fc1_w  = (const float*)d_in[5];
    const float* fc1_b  = (const float*)d_in[6];
    const float* bn1_g  = (const float*)d_in[7];
    const float* bn1_b  = (const float*)d_in[8];
    const float* fc2_w  = (const float*)d_in[9];
    const float* fc2_b  = (const float*)d_in[10];
    const float* bn2_g  = (const float*)d_in[11];
    const float* bn2_b  = (const float*)d_in[12];
    const float* pred_w = (const float*)d_in[13];
    const float* pred_b = (const float*)d_in[14];
    float* out = (float*)d_out;

    const size_t nd = (size_t)N_NODES * DIM;
    float* bufs[3] = { (float*)d_ws, (float*)d_ws + nd, (float*)d_ws + 2 * nd };
    float* stats = (float*)d_ws + 3 * nd;       // 256 floats
    float* ss    = stats + 128;                 // fused scale/shift

    const int n_elems     = N_NODES * DIM;
    const int ew_blocks   = (n_elems + 255) / 256;                       // 25000
    const int sc_blocks   = (int)(((long long)N_EDGES * 4 + 255) / 256); // 25000
    const int gemm_blocks = (N_NODES / 16 + 7) / 8;                      // 782
    const int pool_blocks = (N_NODES + 255) / 256;

    // out = sum_l pred_b[l]
    k_init_out<<<1, N_GRAPHS, 0, stream>>>(pred_b, out);
    // rep 0 pooling (feat)
    k_pool<<<pool_blocks, 256, 0, stream>>>(feat, gids, pred_w, out);

    const float* h = feat;
    int hidx = -1;
    for (int l = 0; l < N_LAYERS; ++l) {
        int a = (hidx < 0) ? 0 : (hidx + 1) % 3;
        int b = (a + 1) % 3;
        int c = (hidx < 0) ? 2 : hidx;          // z2 reuses old-h buffer (dead after scatter)
        float* agg = bufs[a];
        float* z1  = bufs[b];
        float* z2  = bufs[c];

        // agg = (1+eps)*h ; agg[dst] += h[src]
        k_scale<<<ew_blocks, 256, 0, stream>>>(h, agg, eps, l, n_elems);
        k_scatter<<<sc_blocks, 256, 0, stream>>>(h, esrc, edst, agg);

        // z1 = agg @ fc1_w[l] + fc1_b[l]   (raw input, no fused BN)
        k_gemm64<0><<<gemm_blocks, 256, 0, stream>>>(agg, fc1_w + l * DIM * DIM,
                                                     fc1_b + l * DIM, ss, z1);
        // BN1 statistics over z1 -> fused scale/shift
        k_zero_stats<<<1, 128, 0, stream>>>(stats);
        k_colstats<<<512, 256, 0, stream>>>(z1, stats);
        k_bn_finalize<<<1, 64, 0, stream>>>(stats, bn1_g + l * DIM, bn1_b + l * DIM, ss);

        // z2 = relu(BN1(z1)) @ fc2_w[l] + fc2_b[l]   (BN1+ReLU fused into A load)
        k_gemm64<1><<<gemm_blocks, 256, 0, stream>>>(z1, fc2_w + l * DIM * DIM,
                                                     fc2_b + l * DIM, ss, z2);
        // BN2 statistics over z2 -> fused scale/shift, then materialize h
        k_zero_stats<<<1, 128, 0, stream>>>(stats);
        k_colstats<<<512, 256, 0, stream>>>(z2, stats);
        k_bn_finalize<<<1, 64, 0, stream>>>(stats, bn2_g + l * DIM, bn2_b + l * DIM, ss);
        k_bn_relu<<<ew_blocks, 256, 0, stream>>>(z2, ss, n_elems);

        h = z2; hidx = c;
        // rep l+1 pooling
        k_pool<<<pool_blocks, 256, 0, stream>>>(h, gids, pred_w + (l + 1) * DIM, out);
    }
}